// SelectiveStructuredSSM_6459630813399
// MI455X (gfx1250) — compile-verified
//
#include <hip/hip_runtime.h>
#include <hip/hip_bf16.h>

#define BATCH 2
#define LSEQ  4096
#define DDIM  768
#define NSTATE 16

typedef float f32x2 __attribute__((ext_vector_type(2)));
typedef float f32x4 __attribute__((ext_vector_type(4)));
typedef float f32x8 __attribute__((ext_vector_type(8)));

// ---------------------------------------------------------------------------
// Pass 1: Bproj = x@W_B + b_B, Cproj = x@W_C + b_C, s = x@W_D + b_D
// One wave per 16-row tile of x; f32 WMMA 16x16x4, K-loop over D.
// ---------------------------------------------------------------------------
__global__ __launch_bounds__(256) void ssm_proj_kernel(
    const float* __restrict__ x,   // [B*L, D]
    const float* __restrict__ WB,  // [D, N]
    const float* __restrict__ bB,  // [N]
    const float* __restrict__ WC,  // [D, N]
    const float* __restrict__ bC,  // [N]
    const float* __restrict__ WD,  // [D]
    const float* __restrict__ bD,  // [1]
    float* __restrict__ Bproj,     // [B*L, N]
    float* __restrict__ Cproj,     // [B*L, N]
    float* __restrict__ sArr)      // [B*L]
{
    const int lane = threadIdx.x & 31;
    const int wave = threadIdx.x >> 5;
    const int tile = blockIdx.x * 8 + wave;   // 16-row tile of (B*L)
    const int row0 = tile * 16;
    const int n    = lane & 15;               // N column / A row (M)
    const int khi  = (lane >> 4) * 2;         // K sub-offset: 0 or 2

    f32x8 accB = {0.f,0.f,0.f,0.f,0.f,0.f,0.f,0.f};
    f32x8 accC = {0.f,0.f,0.f,0.f,0.f,0.f,0.f,0.f};
    float sacc = 0.f;

    const float* xrow = x + (size_t)(row0 + n) * DDIM + khi;  // A-frag: M=n
    const float* wb   = WB + n;
    const float* wc   = WC + n;

    for (int kk = 0; kk < DDIM; kk += 4) {
        f32x2 a;                       // A: 16x4 fp32 fragment
        a.x = xrow[kk + 0];            // K = kk + khi
        a.y = xrow[kk + 1];            // K = kk + khi + 1
        f32x2 fb, fc;                  // B: 4x16 fp32 fragments
        fb.x = wb[(kk + khi)     * NSTATE];
        fb.y = wb[(kk + khi + 1) * NSTATE];
        fc.x = wc[(kk + khi)     * NSTATE];
        fc.y = wc[(kk + khi + 1) * NSTATE];
        const float wd0 = WD[kk + khi];
        const float wd1 = WD[kk + khi + 1];
        accB = __builtin_amdgcn_wmma_f32_16x16x4_f32(false, a, false, fb,
                                                     (short)0, accB, false, false);
        accC = __builtin_amdgcn_wmma_f32_16x16x4_f32(false, a, false, fc,
                                                     (short)0, accC, false, false);
        sacc = fmaf(a.x, wd0, sacc);   // GEMV with W_D on the same A-frags
        sacc = fmaf(a.y, wd1, sacc);
    }

    // s: combine the two K-halves (lane <-> lane^16), lanes 0..15 hold row n
    sacc += __shfl_xor(sacc, 16, 32);
    if (lane < 16) sArr[row0 + n] = sacc + bD[0];

    const float biasB = bB[n];
    const float biasC = bC[n];
    const int   hi8   = (lane >> 4) * 8;     // C/D: M = v + hi8
#pragma unroll
    for (int v = 0; v < 8; ++v) {
        const int row = row0 + v + hi8;
        Bproj[row * NSTATE + n] = accB[v] + biasB;
        Cproj[row * NSTATE + n] = accC[v] + biasC;
    }
}

// ---------------------------------------------------------------------------
// Pass 2: delta[b,l,d] = softplus(s[b,l] + p_Delta[d])   (s already has b_D)
// ---------------------------------------------------------------------------
__global__ __launch_bounds__(256) void ssm_delta_kernel(
    const float* __restrict__ sArr,    // [B*L]
    const float* __restrict__ pDelta,  // [D]
    float* __restrict__ delta)         // [B*L, D]
{
    const int idx = blockIdx.x * 256 + threadIdx.x;   // < B*L*D
    const int row = idx / DDIM;
    const int d   = idx - row * DDIM;
    const float z = sArr[row] + pDelta[d];
    const float e = __expf(z);
    delta[idx] = (z > 20.f) ? z : __logf(1.f + e);
}

// ---------------------------------------------------------------------------
// Pass 3: sequential scan. 16 lanes per (b,d) pair (n across lanes),
// 2 pairs per wave, 16 pairs per 256-thread block, 96 blocks total.
// 16 time-steps staged through wave-private LDS per iteration (no barriers).
//   a_bar = exp(delta*A_n);  u = (a_bar - A_n)/A_n * B_n * x   (delta cancels)
//   h = a_bar*h + u;  y = sum_n C_n*h  (4-level shfl_xor butterfly)
// ---------------------------------------------------------------------------
__global__ __launch_bounds__(256) void ssm_scan_kernel(
    const float* __restrict__ x,       // [B*L, D]
    const float* __restrict__ delta,   // [B*L, D]
    const float* __restrict__ Bproj,   // [B*L, N]
    const float* __restrict__ Cproj,   // [B*L, N]
    const float* __restrict__ A,       // [D, N]
    float* __restrict__ y)             // [B*L, D]
{
    __shared__ float lds[16 * 544];    // per group: B[16][16] | C[16][16] | xd[16][2]

    const int g = threadIdx.x >> 4;    // group 0..15
    const int n = threadIdx.x & 15;    // state index
    const int p = blockIdx.x * 16 + g; // (b,d) pair
    const int b = p / DDIM;
    const int d = p - b * DDIM;

    float* gB  = lds + g * 544;
    float* gC  = gB + 256;
    float* gXD = gC + 256;

    const float an     = A[d * NSTATE + n];
    const float inv_an = 1.0f / an;
    float h = 0.f;

    const float* Bbase = Bproj + (size_t)b * LSEQ * NSTATE;
    const float* Cbase = Cproj + (size_t)b * LSEQ * NSTATE;
    const float* xcol  = x     + (size_t)b * LSEQ * DDIM + d;
    const float* dcol  = delta + (size_t)b * LSEQ * DDIM + d;
    float*       ycol  = y     + (size_t)b * LSEQ * DDIM + d;

    for (int l0 = 0; l0 < LSEQ; l0 += 16) {
        // --- stage 16 steps of Bproj/Cproj (contiguous 1KB chunks each) ---
        const f32x4* Bc = (const f32x4*)(Bbase + (size_t)l0 * NSTATE);
        const f32x4* Cc = (const f32x4*)(Cbase + (size_t)l0 * NSTATE);
#pragma unroll
        for (int j = 0; j < 4; ++j) {
            ((f32x4*)gB)[j * 16 + n] = Bc[j * 16 + n];
            ((f32x4*)gC)[j * 16 + n] = Cc[j * 16 + n];
        }
        // --- stage x, delta: lane n covers step l0+n ---
        f32x2 xd;
        xd.x = xcol[(size_t)(l0 + n) * DDIM];
        xd.y = dcol[(size_t)(l0 + n) * DDIM];
        ((f32x2*)gXD)[n] = xd;

        float ypack = 0.f;
#pragma unroll
        for (int i = 0; i < 16; ++i) {
            const f32x2 xdel = ((const f32x2*)gXD)[i];   // uniform broadcast
            const float Bn = gB[i * 16 + n];
            const float Cn = gC[i * 16 + n];
            const float abar = __expf(xdel.y * an);
            const float gq   = Bn * xdel.x * inv_an;
            const float u    = (abar - an) * gq;
            h = fmaf(abar, h, u);
            float pr = Cn * h;
            pr += __shfl_xor(pr, 1, 16);
            pr += __shfl_xor(pr, 2, 16);
            pr += __shfl_xor(pr, 4, 16);
            pr += __shfl_xor(pr, 8, 16);
            ypack = (n == i) ? pr : ypack;               // lane n keeps step l0+n
        }
        ycol[(size_t)(l0 + n) * DDIM] = ypack;
    }
}

// ---------------------------------------------------------------------------
extern "C" void kernel_launch(void* const* d_in, const int* in_sizes, int n_in,
                              void* d_out, int out_size, void* d_ws, size_t ws_size,
                              hipStream_t stream) {
    const float* x   = (const float*)d_in[0];   // [B,L,D]
    const float* A   = (const float*)d_in[1];   // [D,N]
    const float* W_B = (const float*)d_in[2];   // [D,N]
    const float* b_B = (const float*)d_in[3];   // [N]
    const float* W_C = (const float*)d_in[4];   // [D,N]
    const float* b_C = (const float*)d_in[5];   // [N]
    const float* W_D = (const float*)d_in[6];   // [D,1]
    const float* b_D = (const float*)d_in[7];   // [1]
    const float* pDl = (const float*)d_in[8];   // [D]
    float* out = (float*)d_out;                 // [B,L,D]

    const size_t rows = (size_t)BATCH * LSEQ;   // 8192
    char* ws = (char*)d_ws;
    float* Bproj = (float*)(ws);                                   // 512 KB
    float* Cproj = (float*)(ws + rows * NSTATE * 4);               // 512 KB
    float* sArr  = (float*)(ws + 2 * rows * NSTATE * 4);           // 32 KB
    float* delta = (float*)(ws + 2 * rows * NSTATE * 4 + rows * 4);// 24 MB

    // Pass 1: WMMA projections (512 tiles, 8 waves/block)
    ssm_proj_kernel<<<dim3((rows / 16) / 8), dim3(256), 0, stream>>>(
        x, W_B, b_B, W_C, b_C, W_D, b_D, Bproj, Cproj, sArr);

    // Pass 2: delta = softplus(s + p_Delta)
    const int total = BATCH * LSEQ * DDIM;
    ssm_delta_kernel<<<dim3(total / 256), dim3(256), 0, stream>>>(sArr, pDl, delta);

    // Pass 3: scan (96 blocks x 256 threads = 1536 (b,d) pairs x 16 lanes)
    ssm_scan_kernel<<<dim3((BATCH * DDIM) / 16), dim3(256), 0, stream>>>(
        x, delta, Bproj, Cproj, A, out);
}